// ModulePopHistory_80487687127448
// MI455X (gfx1250) — compile-verified
//
#include <hip/hip_runtime.h>
#include <hip/hip_bf16.h>
#include <stdint.h>

// EMA-last kernel for MI455X (gfx1250), bandwidth-bound streaming problem.
//   ema[0]=x[0]; ema[j]=0.8*x[j]+0.2*ema[j-1]; out[r]=ema[n_r-1]
// Strategy: one wave32 per row. Row data staged global->LDS with CDNA5 async
// DMA (ASYNCcnt), per-lane EXEC-masked so only ~n elements move. Lanes each
// own 16 contiguous elements, build an affine segment (a,b), do a log2(32)
// shfl_up scan of affine pairs, then the owning lane replays <=16 steps.

#define LMAX 512
#define WPB  8   // waves (=rows) per block

__global__ __launch_bounds__(256) void ema_last_kernel(
    const float* __restrict__ pop,   // [B, L]
    const int*   __restrict__ vlen,  // [B]
    float*       __restrict__ out,   // [B]
    int B, int L)
{
  __shared__ float4 buf[WPB * (LMAX / 4)];   // 8 rows * 2KB = 16KB

  const int lane = threadIdx.x & 31;
  const int wid  = threadIdx.x >> 5;
  const int row  = blockIdx.x * WPB + wid;
  if (row >= B) return;

  const int n = vlen[row];         // uniform per wave, in [1, L]
  const int m = n - 1;             // target index
  const int q = m >> 4;            // lane that owns element m

  // ---- Stage row into LDS via CDNA5 async global->LDS DMA ----
  // One b128 async op moves 32 lanes x 16B = 512B (128 floats).
  const uint64_t gbase = (uint64_t)(uintptr_t)(pop + (size_t)row * (size_t)L);
  const uint32_t lbase = (uint32_t)(uintptr_t)(&buf[wid * (LMAX / 4)]);
#pragma unroll
  for (int c = 0; c < LMAX / 128; ++c) {
    const int elem = c * 128 + lane * 4;      // first of 4 floats this lane moves
    if (elem < n) {                           // per-lane EXEC mask trims traffic
      uint64_t g = gbase + (uint64_t)((unsigned)elem * 4u);
      uint32_t l = lbase + (uint32_t)(elem * 4);
      asm volatile("global_load_async_to_lds_b128 %0, %1, off"
                   :: "v"(l), "v"(g) : "memory");
    }
  }
  asm volatile("s_wait_asynccnt 0" ::: "memory");

  // ---- Each lane reads its 16 contiguous elements from LDS ----
  float x[16];
  const float4* rp = &buf[wid * (LMAX / 4) + lane * 4];
#pragma unroll
  for (int i = 0; i < 4; ++i) {
    float4 v = rp[i];
    x[4 * i + 0] = v.x; x[4 * i + 1] = v.y;
    x[4 * i + 2] = v.z; x[4 * i + 3] = v.w;
  }

  // ---- Local affine composition over the 16-element segment ----
  // f_j(e) = a_j*e + b_j ; a_0 = 0, b_0 = x[0]; else a = 0.2, b = 0.8*x[j].
  const int segStart = lane << 4;
  float A = 1.0f, Bv = 0.0f;
#pragma unroll
  for (int t = 0; t < 16; ++t) {
    const bool first = (segStart + t) == 0;   // only lane 0, t 0
    const float aj = first ? 0.0f : 0.2f;
    const float bj = first ? x[t] : 0.8f * x[t];
    Bv = fmaf(aj, Bv, bj);
    A  = A * aj;
  }

  // ---- Inclusive scan of affine pairs across the wave32 ----
#pragma unroll
  for (int d = 1; d < 32; d <<= 1) {
    float Au = __shfl_up(A, d, 32);
    float Bu = __shfl_up(Bv, d, 32);
    if (lane >= d) {
      Bv = fmaf(A, Bu, Bv);   // F ∘ F_up
      A  = A * Au;
    }
  }

  // Exclusive prefix entering this lane's segment. Since j=0 has a=0, the
  // prefix map is constant: its value is just Bprev.
  float Bprev = __shfl_up(Bv, 1, 32);
  if (lane == 0) Bprev = 0.0f;

  // ---- Owning lane replays <=16 steps to land exactly on index m ----
  if (lane == q) {
    float e = Bprev;
    const int tm = m & 15;
    for (int t = 0; t <= tm; ++t) {
      if (segStart + t == 0) e = x[t];
      else                   e = fmaf(0.2f, e, 0.8f * x[t]);
    }
    out[row] = e;
  }
}

extern "C" void kernel_launch(void* const* d_in, const int* in_sizes, int n_in,
                              void* d_out, int out_size, void* d_ws, size_t ws_size,
                              hipStream_t stream) {
  const float* pop  = (const float*)d_in[0];
  const int*   vlen = (const int*)d_in[1];
  float*       out  = (float*)d_out;

  const int B = in_sizes[1];                 // 65536
  const int L = in_sizes[0] / (B > 0 ? B : 1); // 512

  const int blocks = (B + WPB - 1) / WPB;
  ema_last_kernel<<<blocks, 256, 0, stream>>>(pop, vlen, out, B, L);
}